// HypercubeEmbeddingLayer_893353197937
// MI455X (gfx1250) — compile-verified
//
#include <hip/hip_runtime.h>
#include <hip/hip_bf16.h>

// HypercubeEmbeddingLayer on gfx1250 (MI455X).
//
//   S = X . V^T via V_WMMA_F32_16X16X4_F32 (f32 in / f32 acc, wave32)
//   dist-proxy  = ||v||^2 - 2 x.v   (same argmin as ||x-v||^2)
//   out[token]  = V[argmin]
//
// Work decomposition (latency-bound problem; spread wide):
//   block = 128 threads = 4 wave32s, owns one 16-token tile
//   wave w scans vertex quarter [w*256, (w+1)*256)  -> 16 N-tiles, K=64 each
//   grid  = ntok/16 = 512 blocks -> 2048 waves (fills a ~512-SIMD chip 4-deep)
// Cross-wave argmin combined via LDS with index tie-break (first-min wins,
// matching jnp.argmin).

typedef float v2f __attribute__((ext_vector_type(2)));
typedef float v8f __attribute__((ext_vector_type(8)));

#define EMBED_DIM 64
#define NUM_VERTICES 1024
#define BIGF 3.402823466e38f

// ---------------------------------------------------------------------------
// Kernel 1: per-vertex squared norms -> d_ws (1024 floats)
// ---------------------------------------------------------------------------
__global__ void vertex_norm_kernel(const float* __restrict__ vert,
                                   float* __restrict__ vnorm) {
    int v = blockIdx.x * blockDim.x + threadIdx.x;
    if (v >= NUM_VERTICES) return;
    const float* r = vert + (size_t)v * EMBED_DIM;
    float s = 0.0f;
#pragma unroll
    for (int d = 0; d < EMBED_DIM; ++d) s = fmaf(r[d], r[d], s);
    vnorm[v] = s;
}

// ---------------------------------------------------------------------------
// Kernel 2: WMMA distance GEMM + argmin + vertex gather.
// ---------------------------------------------------------------------------
__global__ __launch_bounds__(128)
void hypercube_wmma_kernel(const int*   __restrict__ ids,
                           const float* __restrict__ raw,
                           const float* __restrict__ vert,
                           const float* __restrict__ vnorm,
                           float*       __restrict__ out,
                           int ntok) {
    const int lane   = threadIdx.x & 31;
    const int wave   = threadIdx.x >> 5;          // 0..3 -> vertex quarter
    const int token0 = blockIdx.x * 16;
    if (token0 >= ntok) return;

    const int row  = lane & 15;              // M (A rows) and N (B cols) slot
    const int koff = (lane >> 4) ? 2 : 0;    // hi half-wave holds K+2,K+3

    // ---- Load A fragments once: 16 tokens x 64 dims, scaled by -2 ---------
    // A 16x4 f32 layout (ISA 7.12.2): lane L -> row M=L&15; VGPR pair holds
    // K = {0,1} (lanes 0-15) / K = {2,3} (lanes 16-31) of each 4-wide K step.
    const int tok = token0 + row;
    const int cid = ids[tok];
    const float* xrow = raw + (size_t)cid * EMBED_DIM + koff;
    v2f a[16];
#pragma unroll
    for (int k = 0; k < 16; ++k) {
        float2 t = *(const float2*)(xrow + 4 * k);
        a[k].x = -2.0f * t.x;
        a[k].y = -2.0f * t.y;
    }

    // Running (min, argmin) for the 8 matrix rows this lane covers.
    float minval[8];
    int   minidx[8];
#pragma unroll
    for (int j = 0; j < 8; ++j) { minval[j] = BIGF; minidx[j] = 0; }

    // ---- Sweep this wave's 16 tiles of 16 vertices -------------------------
    const int nt0 = wave * 16;
    for (int t = 0; t < 16; ++t) {
        const int    n    = (nt0 + t) * 16 + row;
        const float* vrow = vert + (size_t)n * EMBED_DIM + koff;

        v8f c = {};  // accumulates -2 * x.v over K=64 (16 chained WMMAs)
#pragma unroll
        for (int k = 0; k < 16; ++k) {
            float2 t2 = *(const float2*)(vrow + 4 * k);
            v2f b; b.x = t2.x; b.y = t2.y;
            c = __builtin_amdgcn_wmma_f32_16x16x4_f32(
                    /*neg_a=*/false, a[k],
                    /*neg_b=*/false, b,
                    /*c_mod=*/(short)0, c,
                    /*reuse_a=*/false, /*reuse_b=*/false);
        }

        const float vn = vnorm[n];
#pragma unroll
        for (int j = 0; j < 8; ++j) {
            float s = c[j] + vn;            // ||v||^2 - 2 x.v
            bool  p = s < minval[j];        // strict < => first index wins
            minval[j] = p ? s : minval[j];
            minidx[j] = p ? n : minidx[j];
        }
    }

    // ---- Reduce across the 16 columns in each half-wave --------------------
#pragma unroll
    for (int m = 1; m < 16; m <<= 1) {
#pragma unroll
        for (int j = 0; j < 8; ++j) {
            float ov = __shfl_xor(minval[j], m, 32);
            int   oi = __shfl_xor(minidx[j], m, 32);
            bool  p  = (ov < minval[j]) || (ov == minval[j] && oi < minidx[j]);
            minval[j] = p ? ov : minval[j];
            minidx[j] = p ? oi : minidx[j];
        }
    }
    // Lane 0 holds winners for rows 0..7, lane 16 for rows 8..15.

    // ---- Cross-wave combine via LDS ----------------------------------------
    __shared__ float lval[4][16];
    __shared__ int   lidx[4][16];
    __shared__ int   fidx[16];

    if (lane == 0) {
#pragma unroll
        for (int j = 0; j < 8; ++j) { lval[wave][j] = minval[j]; lidx[wave][j] = minidx[j]; }
    }
    if (lane == 16) {
#pragma unroll
        for (int j = 0; j < 8; ++j) { lval[wave][8 + j] = minval[j]; lidx[wave][8 + j] = minidx[j]; }
    }
    __syncthreads();

    if (threadIdx.x < 16) {
        const int r  = threadIdx.x;
        float bv = lval[0][r];
        int   bi = lidx[0][r];
#pragma unroll
        for (int w = 1; w < 4; ++w) {       // ascending w = ascending vertex idx
            float v = lval[w][r];
            int   i = lidx[w][r];
            bool  p = (v < bv) || (v == bv && i < bi);
            bv = p ? v : bv;
            bi = p ? i : bi;
        }
        fidx[r] = bi;
    }
    __syncthreads();

    // ---- Gather winning vertex rows: 8 threads/token, 2 float4 each --------
    {
        const int r = threadIdx.x >> 3;      // token row 0..15
        const int p = threadIdx.x & 7;       // float4 slot 0..7
        const int idx = fidx[r];
        const float4* src = (const float4*)(vert + (size_t)idx * EMBED_DIM);
        float4*       dst = (float4*)(out + (size_t)(token0 + r) * EMBED_DIM);
        dst[p]     = src[p];
        dst[p + 8] = src[p + 8];
    }
}

// ---------------------------------------------------------------------------
extern "C" void kernel_launch(void* const* d_in, const int* in_sizes, int n_in,
                              void* d_out, int out_size, void* d_ws, size_t ws_size,
                              hipStream_t stream) {
    const int*   ids  = (const int*)  d_in[0];   // concept_ids [N]
    const float* raw  = (const float*)d_in[1];   // raw_table   [50257,64]
    const float* vert = (const float*)d_in[2];   // vertex_table[1024,64]
    float*       out  = (float*)d_out;
    float*       vn   = (float*)d_ws;            // 1024 floats scratch

    vertex_norm_kernel<<<(NUM_VERTICES + 255) / 256, 256, 0, stream>>>(vert, vn);

    const int ntok   = in_sizes[0];              // 8192
    const int blocks = (ntok + 15) / 16;         // 16 tokens per block, 4 waves
    hypercube_wmma_kernel<<<blocks, 128, 0, stream>>>(ids, raw, vert, vn, out, ntok);
}